// GRUD_59785944760915
// MI455X (gfx1250) — compile-verified
//
#include <hip/hip_runtime.h>

#define Bn 128
#define Tn 1024
#define Fn 256

typedef __attribute__((ext_vector_type(2))) float v2f;
typedef __attribute__((ext_vector_type(4))) float v4f;
typedef __attribute__((ext_vector_type(8))) float v8f;

__device__ __forceinline__ float fast_sig(float a) {
    // 1 / (1 + e^-a) via native exp + rcp
    return __builtin_amdgcn_rcpf(1.0f + __expf(-a));
}
__device__ __forceinline__ float fast_tanh(float a) {
    // tanh(a) = 2*sigmoid(2a) - 1  (single transcendental)
    return 2.0f * __builtin_amdgcn_rcpf(1.0f + __expf(-2.0f * a)) - 1.0f;
}

// ---------------------------------------------------------------------------
// Kernel 1: elementwise GRU-D recurrence. One (b,f) chain per lane, two
// chains per thread for ILP inside the serial dependence. Coalesced along F.
// ---------------------------------------------------------------------------
__global__ __launch_bounds__(256) void grud_scan(
    const float* __restrict__ inp,   // (B, 3, T, F)
    const float* __restrict__ x_mean,
    const float* __restrict__ w_xz, const float* __restrict__ w_hz, const float* __restrict__ w_mz,
    const float* __restrict__ w_xr, const float* __restrict__ w_hr, const float* __restrict__ w_mr,
    const float* __restrict__ w_xh, const float* __restrict__ w_hh, const float* __restrict__ w_mh,
    const float* __restrict__ b_z,  const float* __restrict__ b_r,  const float* __restrict__ b_h,
    float* __restrict__ hout)        // (B, F)
{
    const int f  = threadIdx.x;
    const int b0 = blockIdx.x;
    const int b1 = blockIdx.x + 64;

    const float xm  = x_mean[f];
    const float wxz = w_xz[f], whz = w_hz[f], wmz = w_mz[f];
    const float wxr = w_xr[f], whr = w_hr[f], wmr = w_mr[f];
    const float wxh = w_xh[f], whh = w_hh[f], wmh = w_mh[f];
    const float bz  = b_z[f],  br  = b_r[f],  bh  = b_h[f];

    const size_t plane = (size_t)Tn * Fn;
    const float* X0 = inp + (size_t)b0 * 3 * plane + f;
    const float* M0 = X0 + plane;
    const float* X1 = inp + (size_t)b1 * 3 * plane + f;
    const float* M1 = X1 + plane;

    float h0 = 0.0f, h1 = 0.0f;
    float x0 = X0[0], m0 = M0[0];
    float x1 = X1[0], m1 = M1[0];

    for (int t = 0; t < Tn; ++t) {
        const int tn = (t + 1 < Tn) ? (t + 1) : t;
        // issue next-step loads before the serial compute chain
        float x0n = X0[(size_t)tn * Fn];
        float m0n = M0[(size_t)tn * Fn];
        float x1n = X1[(size_t)tn * Fn];
        float m1n = M1[(size_t)tn * Fn];

        if (t + 16 < Tn) {  // stream ahead into cache (global_prefetch_b8)
            __builtin_prefetch(X0 + (size_t)(t + 16) * Fn, 0, 3);
            __builtin_prefetch(M0 + (size_t)(t + 16) * Fn, 0, 3);
            __builtin_prefetch(X1 + (size_t)(t + 16) * Fn, 0, 3);
            __builtin_prefetch(M1 + (size_t)(t + 16) * Fn, 0, 3);
        }

        {   // chain 0
            float xi = m0 * x0 + (1.0f - m0) * xm;
            float z  = fast_sig(wxz * xi + whz * h0 + wmz * m0 + bz);
            float r  = fast_sig(wxr * xi + whr * h0 + wmr * m0 + br);
            float ht = fast_tanh(wxh * xi + whh * (r * h0) + wmh * m0 + bh);
            h0 = h0 + z * (ht - h0);       // == (1-z)h + z*ht
        }
        {   // chain 1
            float xi = m1 * x1 + (1.0f - m1) * xm;
            float z  = fast_sig(wxz * xi + whz * h1 + wmz * m1 + bz);
            float r  = fast_sig(wxr * xi + whr * h1 + wmr * m1 + br);
            float ht = fast_tanh(wxh * xi + whh * (r * h1) + wmh * m1 + bh);
            h1 = h1 + z * (ht - h1);
        }
        x0 = x0n; m0 = m0n; x1 = x1n; m1 = m1n;
    }

    hout[b0 * Fn + f] = h0;
    hout[b1 * Fn + f] = h1;
}

// ---------------------------------------------------------------------------
// Kernel 2: classifier  sigmoid(h @ W_cls^T + b_cls)  via V_WMMA_F32_16X16X4_F32.
// One wave per 16-row tile of h. N is padded 2 -> 16 with zero B columns.
//
// Layouts (ISA 7.12.2, 32-bit operands, wave32):
//   A 16x4 : lanes 0-15 hold row M=lane, {v0,v1} = K{0,1}; lanes 16-31: K{2,3}
//   B 4x16 : v0: lanes 0-15 -> K=0,N=lane ; lanes 16-31 -> K=2,N=lane-16
//            v1: lanes 0-15 -> K=1        ; lanes 16-31 -> K=3
//   C/D    : v[r]: lanes 0-15 -> M=r,N=lane ; lanes 16-31 -> M=r+8
// ---------------------------------------------------------------------------
__global__ __launch_bounds__(32) void grud_cls(
    const float* __restrict__ hsrc,   // (B, F)
    const float* __restrict__ Wcls,   // (OUT=2, F)
    const float* __restrict__ bcls,   // (OUT=2)
    float* __restrict__ out)          // (B, OUT)
{
    __shared__ float hs[16 * Fn];     // 16 KB: this tile's 16 rows of h
    __shared__ float ws[16 * Fn];     // 16 KB: W_cls rows 0..1, rows 2..15 zero

    const int lane = threadIdx.x;
    const int r0   = blockIdx.x * 16;

    // stage h tile (contiguous 16*256 floats) into LDS, float4 coalesced
    const v4f* src4 = (const v4f*)(hsrc + (size_t)r0 * Fn);
    v4f* hs4 = (v4f*)hs;
    for (int j = lane; j < (16 * Fn) / 4; j += 32) hs4[j] = src4[j];

    // stage zero-padded W tile
    const v4f* w4 = (const v4f*)Wcls;
    v4f* ws4 = (v4f*)ws;
    for (int j = lane; j < (16 * Fn) / 4; j += 32) {
        v4f v = {};
        if (j < (2 * Fn) / 4) v = w4[j];
        ws4[j] = v;
    }
    __syncthreads();

    const bool hi  = lane >= 16;
    const int  rn  = hi ? lane - 16 : lane;  // A row (M) == B column (N) index
    const int  kof = hi ? 2 : 0;             // this half-wave's K offset

    v8f C = {};
    for (int k = 0; k < Fn; k += 4) {
        v2f A, Bv;
        A.x  = hs[rn * Fn + k + kof];
        A.y  = hs[rn * Fn + k + kof + 1];
        Bv.x = ws[rn * Fn + k + kof];        // zero for N >= 2
        Bv.y = ws[rn * Fn + k + kof + 1];
        C = __builtin_amdgcn_wmma_f32_16x16x4_f32(
                /*neg_a=*/false, A, /*neg_b=*/false, Bv,
                /*c_mod=*/(short)0, C, /*reuse_a=*/false, /*reuse_b=*/false);
    }

    if (rn < 2) {  // lanes 0,1 (M=r) and 16,17 (M=r+8) hold the real columns
        const float bc = bcls[rn];
        #pragma unroll
        for (int rr = 0; rr < 8; ++rr) {
            const int mrow = hi ? rr + 8 : rr;
            const float v  = C[rr] + bc;
            out[(r0 + mrow) * 2 + rn] = 1.0f / (1.0f + __expf(-v));
        }
    }
}

extern "C" void kernel_launch(void* const* d_in, const int* in_sizes, int n_in,
                              void* d_out, int out_size, void* d_ws, size_t ws_size,
                              hipStream_t stream) {
    const float* inp    = (const float*)d_in[0];
    const float* x_mean = (const float*)d_in[1];
    const float* w_xz   = (const float*)d_in[2];
    const float* w_hz   = (const float*)d_in[3];
    const float* w_mz   = (const float*)d_in[4];
    const float* w_xr   = (const float*)d_in[5];
    const float* w_hr   = (const float*)d_in[6];
    const float* w_mr   = (const float*)d_in[7];
    const float* w_xh   = (const float*)d_in[8];
    const float* w_hh   = (const float*)d_in[9];
    const float* w_mh   = (const float*)d_in[10];
    const float* b_z    = (const float*)d_in[11];
    const float* b_r    = (const float*)d_in[12];
    const float* b_h    = (const float*)d_in[13];
    const float* Wcls   = (const float*)d_in[14];
    const float* bcls   = (const float*)d_in[15];

    float* hbuf = (float*)d_ws;  // 128*256 floats = 128 KB scratch for final h

    grud_scan<<<dim3(64), dim3(256), 0, stream>>>(
        inp, x_mean, w_xz, w_hz, w_mz, w_xr, w_hr, w_mr,
        w_xh, w_hh, w_mh, b_z, b_r, b_h, hbuf);

    grud_cls<<<dim3(8), dim3(32), 0, stream>>>(hbuf, Wcls, bcls, (float*)d_out);
}